// KernelConv_32461362823947
// MI455X (gfx1250) — compile-verified
//
#include <hip/hip_runtime.h>
#include <math.h>

// MI455X (gfx1250) implementation.
//
// Strategy: the reference's output is only L=16 floats. All (L,24,N,*) and
// (L,N,*) broadcast-reductions collapse via sum((a-b)^2) = sum(a^2) - 2b*sum(a)
// + N*b^2 into 243 sufficient statistics over N=20000 points:
//   ws[  0..127] Sxn[s*32+d]  = sum_n x_neighbor[n,s,d]
//   ws[128..159] Sxf[d]       = sum_n x_focal[n,d]
//   ws[160..223] Sen[s*16+e]  = sum_n edge_attr_neighbor[n,s,e]
//   ws[224..227] Sa[s]   sum of neighbor intra-angles (cosines)
//   ws[228..231] Sa2[s]  sum of squares
//   ws[232..235] Sl[s]   sum of neighbor lengths
//   ws[236..239] Sl2[s]  sum of squares
//   ws[240] sum x_neighbor^2 (total), ws[241] sum x_focal^2, ws[242] sum edge^2
//
// Phase 1 (reduce_kernel): bandwidth-bound stream (~19 MB @ 23.3 TB/s ~ 1 us)
//   using global_load_b128, wave32 shuffle reduction, LDS combine, f32 atomics.
// Phase 2 (finalize_kernel, 1 wave): 24-perm search from the statistics, then
//   the cross/self dot products (16xK @ Kx16) via V_WMMA_F32_16X16X4_F32:
//   D1 = BestX * broadcast(Sxn)  -> any column = cross dot per row
//   D2 = BestX * BestX^T (Gram)  -> diagonal = self dot per row
//   (f32 A-layout: lanes 0-15 hold K=k0,k0+1; lanes 16-31 hold K=k0+2,k0+3;
//    B mirrors this, so the Gram's B fragment equals the A fragment.)

typedef __attribute__((ext_vector_type(2))) float v2f;
typedef __attribute__((ext_vector_type(8))) float v8f;

#define EPSF 1e-8f
#define HALF_PI 1.57079632679489662f

// itertools.permutations(range(4)) in lexicographic order
__constant__ int PERM[24][4] = {
  {0,1,2,3},{0,1,3,2},{0,2,1,3},{0,2,3,1},{0,3,1,2},{0,3,2,1},
  {1,0,2,3},{1,0,3,2},{1,2,0,3},{1,2,3,0},{1,3,0,2},{1,3,2,0},
  {2,0,1,3},{2,0,3,1},{2,1,0,3},{2,1,3,0},{2,3,0,1},{2,3,1,0},
  {3,0,1,2},{3,0,2,1},{3,1,0,2},{3,1,2,0},{3,2,0,1},{3,2,1,0}
};

__global__ void zero_ws_kernel(float* __restrict__ g) {
  g[threadIdx.x] = 0.0f;  // 256 accumulator slots
}

__device__ inline float wave_red32(float v) {
  v += __shfl_down(v, 16, 32);
  v += __shfl_down(v,  8, 32);
  v += __shfl_down(v,  4, 32);
  v += __shfl_down(v,  2, 32);
  v += __shfl_down(v,  1, 32);
  return v;
}

__global__ __launch_bounds__(256) void reduce_kernel(
    const float* __restrict__ x_focal,     // N,32
    const float* __restrict__ p_focal,     // N,3
    const float* __restrict__ x_neighbor,  // N,4,32
    const float* __restrict__ p_neighbor,  // N,4,3
    const float* __restrict__ edge_nei,    // N,4,16
    float* __restrict__ gacc, int N)
{
  __shared__ float sacc[256];
  const int t = threadIdx.x;
  sacc[t] = 0.0f;
  __syncthreads();
  const int g = blockIdx.x * 256 + t;
  const int stride = gridDim.x * 256;   // multiple of 128 -> fixed column/thread

  // ---- geometric part: angles (cosines) and lengths of p_neighbor - p_focal
  float aSa[4]  = {0,0,0,0}, aSa2[4] = {0,0,0,0};
  float aSl[4]  = {0,0,0,0}, aSl2[4] = {0,0,0,0};
  for (int n = g; n < N; n += stride) {
    const float f0 = p_focal[3*n+0], f1 = p_focal[3*n+1], f2 = p_focal[3*n+2];
    float d[4][3], nr[4];
#pragma unroll
    for (int s = 0; s < 4; s++) {
      d[s][0] = p_neighbor[12*n + 3*s + 0] - f0;
      d[s][1] = p_neighbor[12*n + 3*s + 1] - f1;
      d[s][2] = p_neighbor[12*n + 3*s + 2] - f2;
      nr[s] = sqrtf(d[s][0]*d[s][0] + d[s][1]*d[s][1] + d[s][2]*d[s][2]);
    }
#pragma unroll
    for (int s = 0; s < 4; s++) {
      const int r = (s + 3) & 3;  // roll(p,1)[s] = p[s-1]
      const float dot = d[r][0]*d[s][0] + d[r][1]*d[s][1] + d[r][2]*d[s][2];
      const float c = dot / (fmaxf(nr[r], EPSF) * fmaxf(nr[s], EPSF));
      aSa[s] += c; aSa2[s] += c*c; aSl[s] += nr[s]; aSl2[s] += nr[s]*nr[s];
    }
  }

  // ---- streaming column sums (float4 => global_load_b128)
  const float4* xn4 = (const float4*)x_neighbor;  // N*32 float4s, col = (4g)%128..+3
  float aN[4] = {0,0,0,0}, aN2 = 0.0f;
  const int En = N * 32;
  for (int e = g; e < En; e += stride) {
    const float4 v = xn4[e];
    aN[0]+=v.x; aN[1]+=v.y; aN[2]+=v.z; aN[3]+=v.w;
    aN2 += v.x*v.x + v.y*v.y + v.z*v.z + v.w*v.w;
  }
  const float4* xf4 = (const float4*)x_focal;     // N*8 float4s, col mod 32
  float aF[4] = {0,0,0,0}, aF2 = 0.0f;
  const int Ef = N * 8;
  for (int e = g; e < Ef; e += stride) {
    const float4 v = xf4[e];
    aF[0]+=v.x; aF[1]+=v.y; aF[2]+=v.z; aF[3]+=v.w;
    aF2 += v.x*v.x + v.y*v.y + v.z*v.z + v.w*v.w;
  }
  const float4* ee4 = (const float4*)edge_nei;    // N*16 float4s, col mod 64
  float aE[4] = {0,0,0,0}, aE2 = 0.0f;
  const int Ee = N * 16;
  for (int e = g; e < Ee; e += stride) {
    const float4 v = ee4[e];
    aE[0]+=v.x; aE[1]+=v.y; aE[2]+=v.z; aE[3]+=v.w;
    aE2 += v.x*v.x + v.y*v.y + v.z*v.z + v.w*v.w;
  }

  // ---- combine in LDS (ds_add_f32), then one global atomic per slot per block
  const int cN = (4 * g) & 127;
  const int cF = (4 * g) & 31;
  const int cE = (4 * g) & 63;
#pragma unroll
  for (int j = 0; j < 4; j++) {
    atomicAdd(&sacc[cN + j],       aN[j]);
    atomicAdd(&sacc[128 + cF + j], aF[j]);
    atomicAdd(&sacc[160 + cE + j], aE[j]);
  }
  float sv[19] = { aSa[0],aSa[1],aSa[2],aSa[3], aSa2[0],aSa2[1],aSa2[2],aSa2[3],
                   aSl[0],aSl[1],aSl[2],aSl[3], aSl2[0],aSl2[1],aSl2[2],aSl2[3],
                   aN2, aF2, aE2 };
#pragma unroll
  for (int q = 0; q < 19; q++) {
    const float r = wave_red32(sv[q]);
    if ((t & 31) == 0) atomicAdd(&sacc[224 + q], r);
  }
  __syncthreads();
  if (t < 243) atomicAdd(&gacc[t], sacc[t]);
}

// One 16xK * Kx16 double-product with V_WMMA_F32_16X16X4_F32:
//  cD accumulates A * broadcast(sv): D[m][n] = row_m . sv  (cross term)
//  cG accumulates A * A^T (Gram):    diag   = row_m . row_m (self term)
__device__ inline void wmma_gram_seg(const float* __restrict__ A, int rs, int kk,
                                     const float* __restrict__ sv,
                                     float* __restrict__ dotOut,
                                     float* __restrict__ selfOut, int lane)
{
  v8f cD = {0,0,0,0,0,0,0,0};
  v8f cG = {0,0,0,0,0,0,0,0};
  const int half = (lane < 16) ? 0 : 2;   // f32 A layout K split across lane halves
  const int m = lane & 15;
  for (int k0 = 0; k0 < kk; k0 += 4) {
    v2f a;  a.x  = A[m*rs + k0 + half];   a.y  = A[m*rs + k0 + half + 1];
    v2f bd; bd.x = sv[k0 + half];         bd.y = sv[k0 + half + 1];
    // B fragment layout mirrors A's K split; for the Gram, B = A^T means the
    // fragment this lane supplies (row N=lane%16) equals its A fragment.
    cD = __builtin_amdgcn_wmma_f32_16x16x4_f32(false, a, false, bd, (short)0, cD, false, false);
    cG = __builtin_amdgcn_wmma_f32_16x16x4_f32(false, a, false, a,  (short)0, cG, false, false);
  }
  float td[8], tg[8];
#pragma unroll
  for (int r = 0; r < 8; r++) { td[r] = cD[r]; tg[r] = cG[r]; }
  // C/D layout: vgpr r, lanes 0-15 -> (M=r, N=lane); lanes 16-31 -> (M=r+8, N=lane-16)
  if (lane == 0) {
#pragma unroll
    for (int r = 0; r < 8; r++) dotOut[r] = td[r];        // dot_M independent of N
  }
  if (lane == 16) {
#pragma unroll
    for (int r = 0; r < 8; r++) dotOut[8 + r] = td[r];
  }
  if (lane < 8)   selfOut[lane]      = tg[lane];          // diag M=N=lane
  if (lane >= 24) selfOut[lane - 16] = tg[lane - 24];     // diag M=N=lane-16
}

__global__ __launch_bounds__(32) void finalize_kernel(
    const float* __restrict__ p_support,  // L,4,3
    const float* __restrict__ x_support,  // L,4,32
    const float* __restrict__ e_support,  // L,4,16
    const float* __restrict__ x_center,   // L,1,32
    const float* __restrict__ gacc,
    float* __restrict__ out, int N)
{
  const int lane = threadIdx.x;
  __shared__ float sBX[16*128];
  __shared__ float sBE[16*64];
  __shared__ float sXC[16*32];
  __shared__ float sSxn[128];
  __shared__ float sSxf[32];
  __shared__ float sSen[64];
  __shared__ int   sBidx[16];
  __shared__ float sDotA[16], sSelfA[16], sDotE[16], sSelfE[16], sDotC[16], sSelfC[16];

  for (int i = lane; i < 128; i += 32) sSxn[i] = gacc[i];
  sSxf[lane] = gacc[128 + lane];
  for (int i = lane; i < 64; i += 32) sSen[i] = gacc[160 + i];

  float angle_min = 0.0f, length_sum = 0.0f;
  const float Nf = (float)N;
  if (lane < 16) {
    const int l = lane;
    float ps[4][3], nr[4];
#pragma unroll
    for (int s = 0; s < 4; s++) {
      ps[s][0] = p_support[(l*4+s)*3 + 0];
      ps[s][1] = p_support[(l*4+s)*3 + 1];
      ps[s][2] = p_support[(l*4+s)*3 + 2];
      nr[s] = sqrtf(ps[s][0]*ps[s][0] + ps[s][1]*ps[s][1] + ps[s][2]*ps[s][2]);
    }
    float Sa[4], Sa2[4], Sl[4], Sl2[4];
#pragma unroll
    for (int s = 0; s < 4; s++) {
      Sa[s]  = gacc[224 + s];
      Sa2[s] = gacc[228 + s];
      Sl[s]  = gacc[232 + s];
      Sl2[s] = gacc[236 + s];
    }
    // argmax of arctan(1/x) == first argmin of x (x > 0)
    int best = 0; float bestSum = 3.4e38f;
    for (int p = 0; p < 24; p++) {
      float sum = 0.0f;
#pragma unroll
      for (int s = 0; s < 4; s++) {
        const int r0 = PERM[p][(s + 3) & 3], r1 = PERM[p][s];
        const float dot = ps[r0][0]*ps[r1][0] + ps[r0][1]*ps[r1][1] + ps[r0][2]*ps[r1][2];
        const float b = dot / (fmaxf(nr[r0], EPSF) * fmaxf(nr[r1], EPSF));
        sum += Sa2[s] - 2.0f*b*Sa[s] + Nf*b*b;
      }
      if (sum < bestSum) { bestSum = sum; best = p; }
    }
    angle_min = bestSum;
#pragma unroll
    for (int s = 0; s < 4; s++) {
      const float ls = nr[PERM[best][s]];
      length_sum += Sl2[s] - 2.0f*ls*Sl[s] + Nf*ls*ls;
    }
    sBidx[l] = best;
  }
  __syncthreads();

  // gather best-permutation rows into LDS matrices (16 x K, row-major)
  for (int idx = lane; idx < 16*128; idx += 32) {
    const int row = idx >> 7, k = idx & 127, s = k >> 5, dd = k & 31;
    sBX[idx] = x_support[(row*4 + PERM[sBidx[row]][s])*32 + dd];
  }
  for (int idx = lane; idx < 16*64; idx += 32) {
    const int row = idx >> 6, k = idx & 63, s = k >> 4, ee = k & 15;
    sBE[idx] = e_support[(row*4 + PERM[sBidx[row]][s])*16 + ee];
  }
  for (int idx = lane; idx < 16*32; idx += 32) sXC[idx] = x_center[idx];
  __syncthreads();

  wmma_gram_seg(sBX, 128, 128, sSxn, sDotA, sSelfA, lane);
  wmma_gram_seg(sBE,  64,  64, sSen, sDotE, sSelfE, lane);
  wmma_gram_seg(sXC,  32,  32, sSxf, sDotC, sSelfC, lane);
  __syncthreads();

  if (lane < 16) {
    const float suppSum = gacc[240] - 2.0f*sDotA[lane] + Nf*sSelfA[lane];
    const float centSum = gacc[241] - 2.0f*sDotC[lane] + Nf*sSelfC[lane];
    const float edgeSum = gacc[242] - 2.0f*sDotE[lane] + Nf*sSelfE[lane];
    const float s1 = atanf(1.0f/length_sum) - HALF_PI;
    const float s2 = atanf(1.0f/angle_min)  - HALF_PI;
    const float s3 = atanf(1.0f/suppSum)    - HALF_PI;
    const float s4 = atanf(1.0f/centSum)    - HALF_PI;
    const float s5 = atanf(1.0f/edgeSum)    - HALF_PI;
    const float total = s1*s1 + s2*s2 + s3*s3 + s4*s4 + s5*s5;
    out[lane] = atanf(1.0f/total);
  }
}

extern "C" void kernel_launch(void* const* d_in, const int* in_sizes, int n_in,
                              void* d_out, int out_size, void* d_ws, size_t ws_size,
                              hipStream_t stream) {
  (void)n_in; (void)out_size; (void)ws_size;
  const float* x_focal    = (const float*)d_in[0];
  const float* p_focal    = (const float*)d_in[1];
  const float* x_neighbor = (const float*)d_in[2];
  const float* p_neighbor = (const float*)d_in[3];
  const float* edge_nei   = (const float*)d_in[4];
  const float* x_center   = (const float*)d_in[5];
  const float* x_support  = (const float*)d_in[6];
  const float* e_support  = (const float*)d_in[7];
  const float* p_support  = (const float*)d_in[8];
  float* out  = (float*)d_out;
  float* gacc = (float*)d_ws;  // 256 floats of accumulators
  const int N = in_sizes[1] / 3;   // 20000

  zero_ws_kernel<<<1, 256, 0, stream>>>(gacc);
  reduce_kernel<<<256, 256, 0, stream>>>(x_focal, p_focal, x_neighbor,
                                         p_neighbor, edge_nei, gacc, N);
  finalize_kernel<<<1, 32, 0, stream>>>(p_support, x_support, e_support,
                                        x_center, gacc, out, N);
}